// Laplacian_5660766896726
// MI455X (gfx1250) — compile-verified
//
#include <hip/hip_runtime.h>
#include <math.h>

// ---------------------------------------------------------------------------
// Graph Laplacian diffusion (memory-bound gather/scatter; no GEMM structure).
// CDNA5 paths used: wave32 lane==feature mapping (F=32), async global->LDS
// staging of edge metadata (ASYNCcnt), coalesced 128B atomics via [N,32]
// accumulator layout. All data is L2-resident on MI455X (192MB L2).
// ---------------------------------------------------------------------------

#define DEV __device__ __forceinline__

#if defined(__has_builtin)
#if __has_builtin(__builtin_amdgcn_global_load_async_to_lds_b128) && \
    __has_builtin(__builtin_amdgcn_s_wait_asynccnt)
#define USE_ASYNC_BUILTIN 1
#endif
#endif
#ifndef USE_ASYNC_BUILTIN
#define USE_ASYNC_BUILTIN 0
#endif

// GCC-style vector to exactly match the builtin's V4i parameter type.
typedef int lap_v4i __attribute__((vector_size(16)));
typedef __attribute__((address_space(1))) lap_v4i* lap_gv4i_p;
typedef __attribute__((address_space(3))) lap_v4i* lap_lv4i_p;

DEV void lds_copy_b128_async(void* lds, const void* gmem) {
#if USE_ASYNC_BUILTIN
  __builtin_amdgcn_global_load_async_to_lds_b128(
      (lap_gv4i_p)((void*)gmem), (lap_lv4i_p)(lds),
      /*offset=*/0, /*cpol=*/0);
#else
  unsigned lds_off = (unsigned)(unsigned long long)lds;  // low 32 bits = LDS addr
  asm volatile("global_load_async_to_lds_b128 %0, %1, off"
               :
               : "v"(lds_off), "v"(gmem)
               : "memory");
#endif
}

DEV void wait_async_zero() {
#if USE_ASYNC_BUILTIN
  __builtin_amdgcn_s_wait_asynccnt(0);
#else
  asm volatile("s_wait_asynccnt 0x0" ::: "memory");
#endif
}

// ---------------------------------------------------------------------------
// Kernel A: zero the [N,32] accumulator, zero deg.
__global__ __launch_bounds__(256) void lap_init_kernel(float* __restrict__ deg,
                                                       float* __restrict__ lapt,
                                                       int N) {
  int t = blockIdx.x * blockDim.x + threadIdx.x;
  if (t < N * 32) lapt[t] = 0.0f;
  if (t < N) deg[t] = 0.0f;
}

// Kernel B: d_full[e] = sigmoid(0.5*(w[perm[e]] + w[e])) for e<E0, 1.0 for
// trailing self-loops; scatter-add onto deg[dst[e]].
__global__ __launch_bounds__(256) void lap_weight_deg_kernel(
    const float* __restrict__ w, const int* __restrict__ perm,
    const int* __restrict__ dst, float* __restrict__ dfull,
    float* __restrict__ deg, int E0, int E) {
  int e = blockIdx.x * blockDim.x + threadIdx.x;
  if (e >= E) return;
  float d;
  if (e < E0) {
    float x = 0.5f * (w[perm[e]] + w[e]);
    d = 1.0f / (1.0f + expf(-x));
    dfull[e] = d;
  } else {
    d = 1.0f;  // self-loop fixed weight
  }
  atomicAdd(&deg[dst[e]], d);
}

// Kernel C: per-node 1/sqrt(deg) and 1/deg (deg >= 1 due to self loops).
__global__ __launch_bounds__(256) void lap_node_norm_kernel(
    const float* __restrict__ deg, float* __restrict__ rsd,
    float* __restrict__ invd, int N) {
  int n = blockIdx.x * blockDim.x + threadIdx.x;
  if (n >= N) return;
  float dv = deg[n];
  rsd[n] = 1.0f / sqrtf(dv);
  invd[n] = 1.0f / dv;
}

// Kernel D: transpose y [32,N] -> y_t [N,32] so per-edge feature gathers are
// one contiguous 128B line. Block (32,8), 32x32 LDS tile.
__global__ __launch_bounds__(256) void lap_t_FN_to_NF_kernel(
    const float* __restrict__ in, float* __restrict__ out, int N) {
  __shared__ float tile[32][33];
  int n0 = blockIdx.x * 32;
  int tx = threadIdx.x, ty = threadIdx.y;
  {
    int n = n0 + tx;
    if (n < N) {
      for (int r = 0; r < 4; ++r) {
        int f = ty + r * 8;
        tile[f][tx] = in[f * N + n];  // coalesced along n
      }
    }
  }
  __syncthreads();
  for (int r = 0; r < 4; ++r) {
    int j = ty + r * 8;
    int n = n0 + j;
    if (n < N) out[n * 32 + tx] = tile[tx][j];  // coalesced along f
  }
}

// Kernel F: transpose lap_t [N,32] -> out [32,N] (LAMB==1).
__global__ __launch_bounds__(256) void lap_t_NF_to_FN_kernel(
    const float* __restrict__ in, float* __restrict__ out, int N) {
  __shared__ float tile[32][33];
  int n0 = blockIdx.x * 32;
  int tx = threadIdx.x, ty = threadIdx.y;
  for (int r = 0; r < 4; ++r) {
    int j = ty + r * 8;
    int n = n0 + j;
    if (n < N) tile[j][tx] = in[n * 32 + tx];  // coalesced along f
  }
  __syncthreads();
  int n = n0 + tx;
  if (n < N) {
    for (int r = 0; r < 4; ++r) {
      int f = ty + r * 8;
      out[f * N + n] = tile[tx][f];  // coalesced along n
    }
  }
}

// Kernel E: per-edge messages. One wave32 per edge, lane == feature.
// Edge metadata (src,dst,d) staged per 2048-edge chunk into LDS via
// async global->LDS b128 copies (CDNA5 ASYNCcnt path).
// Self-loop tail edges are skipped entirely (their message is exactly 0).
#define LAP_CHUNK 2048

__global__ __launch_bounds__(256) void lap_edge_msg_kernel(
    const int* __restrict__ src, const int* __restrict__ dst,
    const float* __restrict__ dfull, const float* __restrict__ rsd,
    const float* __restrict__ invd, const float* __restrict__ yt,
    float* __restrict__ lapt, int E0) {
  __shared__ __attribute__((aligned(16))) int s_src[LAP_CHUNK];
  __shared__ __attribute__((aligned(16))) int s_dst[LAP_CHUNK];
  __shared__ __attribute__((aligned(16))) float s_d[LAP_CHUNK];

  int base = blockIdx.x * LAP_CHUNK;
  if (base >= E0) return;
  int len = E0 - base;
  if (len > LAP_CHUNK) len = LAP_CHUNK;
  int tid = threadIdx.x;

  if (base + LAP_CHUNK <= E0) {
    // Full chunk: async 128-bit DMA into LDS (2 x b128 per array per thread).
    for (int i = tid; i < LAP_CHUNK / 4; i += 256) {
      lds_copy_b128_async(&s_src[i * 4], src + base + i * 4);
      lds_copy_b128_async(&s_dst[i * 4], dst + base + i * 4);
      lds_copy_b128_async(&s_d[i * 4], dfull + base + i * 4);
    }
    wait_async_zero();  // each wave drains its own ASYNCcnt before barrier
  } else {
    for (int i = tid; i < len; i += 256) {
      s_src[i] = src[base + i];
      s_dst[i] = dst[base + i];
      s_d[i] = dfull[base + i];
    }
  }
  __syncthreads();

  int wave = tid >> 5;
  int lane = tid & 31;  // lane == feature index (F == 32 == wave32)
  for (int j = wave; j < len; j += 8) {
    int es = s_src[j];       // wave-uniform LDS broadcast
    int ed = s_dst[j];
    float d = s_d[j];
    float a = d * rsd[es] * rsd[ed];  // d / sqrt(deg_s * deg_d)
    float b = d * invd[ed];           // d / deg_d
    float ys = yt[es * 32 + lane];    // one 128B line per wave
    float yd = yt[ed * 32 + lane];    // one 128B line per wave
    // coalesced 128B atomic scatter onto destination node
    atomicAdd(&lapt[ed * 32 + lane], fmaf(a, ys, -(b * yd)));
  }
}

// ---------------------------------------------------------------------------
extern "C" void kernel_launch(void* const* d_in, const int* in_sizes, int n_in,
                              void* d_out, int out_size, void* d_ws,
                              size_t ws_size, hipStream_t stream) {
  const float* y = (const float*)d_in[0];       // [32, N]
  const float* dist_w = (const float*)d_in[1];  // [E0]
  const int* edge_index = (const int*)d_in[2];  // [2, E]
  const int* perm = (const int*)d_in[3];        // [E0]

  const int F = 32;
  const int N = in_sizes[0] / F;
  const int E0 = in_sizes[1];
  const int E = in_sizes[2] / 2;

  const int* src = edge_index;
  const int* dst = edge_index + E;

  // Workspace layout (256B-aligned slabs).
  char* ws = (char*)d_ws;
  size_t off = 0;
  auto slab = [&](size_t nfloats) {
    size_t r = off;
    off += ((nfloats * sizeof(float)) + 255) & ~(size_t)255;
    return r;
  };
  float* deg   = (float*)(ws + slab((size_t)N));
  float* rsd   = (float*)(ws + slab((size_t)N));
  float* invd  = (float*)(ws + slab((size_t)N));
  float* dfull = (float*)(ws + slab((size_t)E0));
  float* yt    = (float*)(ws + slab((size_t)N * F));
  float* lapt  = (float*)(ws + slab((size_t)N * F));
  (void)ws_size;
  (void)n_in;
  (void)out_size;

  float* out = (float*)d_out;  // [32, N]

  // A: init accumulator + deg
  {
    int total = N * F;
    int blocks = (total + 255) / 256;
    lap_init_kernel<<<blocks, 256, 0, stream>>>(deg, lapt, N);
  }
  // B: edge weights + degree scatter
  {
    int blocks = (E + 255) / 256;
    lap_weight_deg_kernel<<<blocks, 256, 0, stream>>>(dist_w, perm, dst, dfull,
                                                      deg, E0, E);
  }
  // C: per-node normalizers
  {
    int blocks = (N + 255) / 256;
    lap_node_norm_kernel<<<blocks, 256, 0, stream>>>(deg, rsd, invd, N);
  }
  // D: y [F,N] -> y_t [N,F]
  {
    int blocks = (N + 31) / 32;
    lap_t_FN_to_NF_kernel<<<blocks, dim3(32, 8, 1), 0, stream>>>(y, yt, N);
  }
  // E: edge messages (only real edges; self-loop tail contributes exactly 0)
  {
    int blocks = (E0 + LAP_CHUNK - 1) / LAP_CHUNK;
    lap_edge_msg_kernel<<<blocks, 256, 0, stream>>>(src, dst, dfull, rsd, invd,
                                                    yt, lapt, E0);
  }
  // F: lap_t [N,F] -> out [F,N] (writes every output element)
  {
    int blocks = (N + 31) / 32;
    lap_t_NF_to_FN_kernel<<<blocks, dim3(32, 8, 1), 0, stream>>>(lapt, out, N);
  }
}